// SpatialCrossAttention_18545668784593
// MI455X (gfx1250) — compile-verified
//
#include <hip/hip_runtime.h>
#include <math.h>

typedef __attribute__((ext_vector_type(2))) float v2f;
typedef __attribute__((ext_vector_type(8))) float v8f;

#define NCAMS 6
#define NQ    1600
#define EMBD  256
#define NHEAD 8
#define HD    32
#define KT    32
#define NKEY  5440

__device__ __constant__ int d_lvlH[4] = {64, 32, 16, 8};
__device__ __constant__ int d_lvlW[4] = {64, 32, 16, 8};
__device__ __constant__ int d_lvlS[4] = {0, 4096, 5120, 5376};

__device__ __forceinline__ v8f wmma4(v2f a, v2f b, v8f c) {
  // D = A(16x4 f32) * B(4x16 f32) + C(16x16 f32)
  return __builtin_amdgcn_wmma_f32_16x16x4_f32(false, a, false, b, (short)0, c,
                                               false, false);
}

__device__ __forceinline__ v2f ld2(const float* p) {
  return *(const v2f*)p;  // 8B-aligned by construction -> global_load_b64
}

__device__ __forceinline__ void bilin_setup(float gx, float gy, int W, int H,
                                            int* idx, float* w) {
  float fx = (gx + 1.0f) * (W * 0.5f) - 0.5f;
  float fy = (gy + 1.0f) * (H * 0.5f) - 0.5f;
  float x0f = floorf(fx), y0f = floorf(fy);
  float wx1 = fx - x0f, wy1 = fy - y0f;
  int x0 = (int)x0f, y0 = (int)y0f;
#pragma unroll
  for (int i = 0; i < 4; ++i) {
    int xi = x0 + (i & 1);
    int yi = y0 + ((i >> 1) & 1);
    bool inb = (xi >= 0) && (xi < W) && (yi >= 0) && (yi < H);
    int xc = min(max(xi, 0), W - 1);
    int yc = min(max(yi, 0), H - 1);
    idx[i] = yc * W + xc;
    float wx = (i & 1) ? wx1 : (1.0f - wx1);
    float wy = (i & 2) ? wy1 : (1.0f - wy1);
    w[i] = inb ? (wx * wy) : 0.0f;
  }
}

// ---------------- K-1: transpose a weight matrix [K,N] -> [N,K] ------------
__global__ void k_transpose(const float* __restrict__ src,
                            float* __restrict__ dst, int K, int N) {
  int i = blockIdx.x * blockDim.x + threadIdx.x;
  if (i >= K * N) return;
  int k = i / N, n = i - k * N;
  dst[n * K + k] = src[i];
}

// ---------------- K0a: init first-valid sentinels ----------------
__global__ void k_init(int* q0) {
  int t = threadIdx.x;
  if (t < NCAMS) q0[t] = NQ;
}

// ---------------- K0: valid mask, per-query cam count, first valid per cam --
__global__ void k_mask(const int* __restrict__ bev, float* __restrict__ validf,
                       float* __restrict__ countq, int* __restrict__ q0) {
  int q = blockIdx.x * blockDim.x + threadIdx.x;
  if (q >= NQ) return;
  int cnt = 0;
#pragma unroll
  for (int c = 0; c < NCAMS; ++c) {
    int s = 0;
#pragma unroll
    for (int z = 0; z < 4; ++z) s += bev[(c * NQ + q) * 4 + z];
    int v = (s > 0) ? 1 : 0;
    validf[c * NQ + q] = (float)v;
    cnt += v;
    if (v) atomicMin(&q0[c], q);
  }
  countq[q] = (float)(cnt < 1 ? 1 : cnt);
}

// ---------------- K1: fused [Wq|Woff] projection (WMMA) ----------------
// per (cam, head): A[1600,32] x B[32,96] -> qproj[..,32], offw[..,64]
// WqT: [32,32] (N,K), WoffT: [64,32] (N,K)
__global__ void k_qoff(const float* __restrict__ query,
                       const float* __restrict__ validf,
                       const float* __restrict__ WqT, const float* __restrict__ bq,
                       const float* __restrict__ WoffT,
                       const float* __restrict__ boff,
                       float* __restrict__ qproj, float* __restrict__ offw) {
  int c = blockIdx.z, hd = blockIdx.y;
  int mb = blockIdx.x * 16;
  int lane = threadIdx.x;
  int m = lane & 15, hi = lane >> 4, ko = hi * 2, n = lane & 15;
  int q = mb + m;
  float vm = validf[c * NQ + q];
  v2f A[8];
#pragma unroll
  for (int s = 0; s < 8; ++s)
    A[s] = ld2(&query[q * EMBD + hd * HD + s * 4 + ko]) * vm;
#pragma unroll
  for (int nt = 0; nt < 6; ++nt) {
    v8f acc = {0.f, 0.f, 0.f, 0.f, 0.f, 0.f, 0.f, 0.f};
    int jb = nt * 16;
    const float* BT = (jb < 32) ? &WqT[(jb + n) * HD]          // wave-uniform
                                : &WoffT[(jb - 32 + n) * HD];
#pragma unroll
    for (int s = 0; s < 8; ++s)
      acc = wmma4(A[s], ld2(&BT[s * 4 + ko]), acc);
    int j = jb + n;
    float bias = (j < 32) ? bq[j] : boff[j - 32];
#pragma unroll
    for (int r = 0; r < 8; ++r) {
      int qq = mb + r + hi * 8;
      float val = acc[r] + bias;
      if (j < 32)
        qproj[(((size_t)c * NQ + qq) * NHEAD + hd) * HD + j] = val;
      else
        offw[(((size_t)c * NQ + qq) * NHEAD + hd) * 64 + (j - 32)] = val;
    }
  }
}

// ---------------- K2: key sampling (first valid query grid) + Wk proj -------
__global__ void k_keys(const float* __restrict__ key,
                       const float* __restrict__ refcam,
                       const float* __restrict__ validf,
                       const int* __restrict__ q0,
                       const float* __restrict__ offw,
                       const float* __restrict__ WkT, const float* __restrict__ bk,
                       float* __restrict__ kproj) {
  int hd = blockIdx.x, c = blockIdx.y;
  int t = threadIdx.x;  // 32 threads, t == kk
  __shared__ float ks[KT][HD + 2];  // stride 34 keeps rows 8B-aligned
  int j0 = q0[c];
  if (j0 >= NQ) j0 = 0;
  float vm = validf[c * NQ + j0];
  int kk = t;
  int lvl = kk >> 3, pz = kk & 7, z = pz & 3;
  int W = d_lvlW[lvl], H = d_lvlH[lvl], st = d_lvlS[lvl];
  float rx = refcam[((size_t)c * NQ + j0) * 8 + z * 2 + 0] * vm;
  float ry = refcam[((size_t)c * NQ + j0) * 8 + z * 2 + 1] * vm;
  const float* ob = &offw[(((size_t)c * NQ + j0) * NHEAD + hd) * 64 + lvl * 16 + pz * 2];
  float gx = 2.0f * (rx + ob[0] / (float)W) - 1.0f;
  float gy = 2.0f * (ry + ob[1] / (float)H) - 1.0f;
  int idx[4];
  float w[4];
  bilin_setup(gx, gy, W, H, idx, w);
  const float* kb_ = &key[(size_t)c * NKEY * EMBD + hd * HD];
  for (int dch = 0; dch < HD; ++dch) {
    float v = 0.f;
#pragma unroll
    for (int i = 0; i < 4; ++i)
      v += w[i] * kb_[(size_t)(st + idx[i]) * EMBD + dch];
    ks[kk][dch] = v;
  }
  __syncthreads();
  // kproj[c][hd][32][32] = ks[32,32] @ Wk[32,32] + bk  (WMMA)
  int lane = t, m = lane & 15, hi = lane >> 4, ko = hi * 2, n = lane & 15;
#pragma unroll
  for (int mt = 0; mt < 2; ++mt) {
#pragma unroll
    for (int nt = 0; nt < 2; ++nt) {
      v8f acc = {0.f, 0.f, 0.f, 0.f, 0.f, 0.f, 0.f, 0.f};
#pragma unroll
      for (int s = 0; s < 8; ++s) {
        v2f A = *(const v2f*)&ks[mt * 16 + m][s * 4 + ko];
        v2f B = ld2(&WkT[(nt * 16 + n) * HD + s * 4 + ko]);
        acc = wmma4(A, B, acc);
      }
      int j = nt * 16 + n;
#pragma unroll
      for (int r = 0; r < 8; ++r) {
        int row = mt * 16 + r + hi * 8;
        kproj[(((size_t)c * NHEAD + hd) * KT + row) * HD + j] = acc[r] + bk[j];
      }
    }
  }
}

// ---------------- K3: QK scores (WMMA) + softmax ----------------
__global__ void k_scores(const float* __restrict__ qproj,
                         const float* __restrict__ kproj,
                         float* __restrict__ attw) {
  int c = blockIdx.z, hd = blockIdx.y;
  int mb = blockIdx.x * 16;
  int lane = threadIdx.x;
  __shared__ float sc[16][KT];
  int m = lane & 15, hi = lane >> 4, ko = hi * 2, n = lane & 15;
  v2f A[8];
#pragma unroll
  for (int s = 0; s < 8; ++s)
    A[s] = ld2(&qproj[(((size_t)c * NQ + mb + m) * NHEAD + hd) * HD + s * 4 + ko]);
#pragma unroll
  for (int nt = 0; nt < 2; ++nt) {
    v8f acc = {0.f, 0.f, 0.f, 0.f, 0.f, 0.f, 0.f, 0.f};
#pragma unroll
    for (int s = 0; s < 8; ++s) {
      // B[k][n] = kproj[key n][dim k]: kproj is already N-major, K-adjacent
      v2f B = ld2(&kproj[(((size_t)c * NHEAD + hd) * KT + nt * 16 + n) * HD + s * 4 + ko]);
      acc = wmma4(A[s], B, acc);
    }
#pragma unroll
    for (int r = 0; r < 8; ++r) sc[r + hi * 8][nt * 16 + n] = acc[r];
  }
  __syncthreads();
  if (lane < 16) {
    int row = lane;
    float mx = -1e30f;
#pragma unroll
    for (int k = 0; k < KT; ++k) mx = fmaxf(mx, sc[row][k]);
    float ssum = 0.f;
#pragma unroll
    for (int k = 0; k < KT; ++k) ssum += __expf(sc[row][k] - mx);
    float inv = 1.f / ssum;
#pragma unroll
    for (int k = 0; k < KT; ++k)
      attw[(((size_t)c * NQ + mb + row) * NHEAD + hd) * KT + k] =
          __expf(sc[row][k] - mx) * inv;
  }
}

// ---------------- K4: value bilinear gather + attw-weighted sum ----------
// block = (q, c); 8 waves = heads; lane = channel (coalesced 128B per wave)
__global__ void k_vsamp(const float* __restrict__ value,
                        const float* __restrict__ refcam,
                        const float* __restrict__ validf,
                        const float* __restrict__ offw,
                        const float* __restrict__ attw,
                        float* __restrict__ vsum) {
  int q = blockIdx.x, c = blockIdx.y;
  int hd = threadIdx.x >> 5;
  int dch = threadIdx.x & 31;
  float* outp = &vsum[(((size_t)c * NQ + q) * NHEAD + hd) * HD + dch];
  float vm = validf[c * NQ + q];
  if (vm == 0.f) {
    *outp = 0.f;
    return;
  }
  const float* ob = &offw[(((size_t)c * NQ + q) * NHEAD + hd) * 64];
  const float* aw = &attw[(((size_t)c * NQ + q) * NHEAD + hd) * KT];
  const float* vb = &value[(size_t)c * NKEY * EMBD + hd * HD + dch];
  float rx[4], ry[4];
#pragma unroll
  for (int z = 0; z < 4; ++z) {
    rx[z] = refcam[((size_t)c * NQ + q) * 8 + z * 2 + 0];
    ry[z] = refcam[((size_t)c * NQ + q) * 8 + z * 2 + 1];
  }
  float acc = 0.f;
#pragma unroll 4
  for (int kk = 0; kk < KT; ++kk) {
    int lvl = kk >> 3, pz = kk & 7, z = pz & 3;
    int W = d_lvlW[lvl], H = d_lvlH[lvl], st = d_lvlS[lvl];
    float gx = 2.0f * (rx[z] + ob[lvl * 16 + pz * 2 + 0] / (float)W) - 1.0f;
    float gy = 2.0f * (ry[z] + ob[lvl * 16 + pz * 2 + 1] / (float)H) - 1.0f;
    int idx[4];
    float w[4];
    bilin_setup(gx, gy, W, H, idx, w);
    float v = 0.f;
#pragma unroll
    for (int i = 0; i < 4; ++i)
      v += w[i] * vb[(size_t)(st + idx[i]) * EMBD];
    acc = fmaf(aw[kk], v, acc);
  }
  *outp = acc;
}

// ---------------- K5: Wv projection (WMMA) + mask ----------------
__global__ void k_vproj(const float* __restrict__ vsum,
                        const float* __restrict__ WvT, const float* __restrict__ bv,
                        const float* __restrict__ validf,
                        float* __restrict__ attF) {
  int c = blockIdx.z, hd = blockIdx.y, mb = blockIdx.x * 16;
  int lane = threadIdx.x;
  int m = lane & 15, hi = lane >> 4, ko = hi * 2, n = lane & 15;
  v2f A[8];
#pragma unroll
  for (int s = 0; s < 8; ++s)
    A[s] = ld2(&vsum[(((size_t)c * NQ + mb + m) * NHEAD + hd) * HD + s * 4 + ko]);
#pragma unroll
  for (int nt = 0; nt < 2; ++nt) {
    v8f acc = {0.f, 0.f, 0.f, 0.f, 0.f, 0.f, 0.f, 0.f};
#pragma unroll
    for (int s = 0; s < 8; ++s)
      acc = wmma4(A[s], ld2(&WvT[(nt * 16 + n) * HD + s * 4 + ko]), acc);
    int j = nt * 16 + n;
#pragma unroll
    for (int r = 0; r < 8; ++r) {
      int qq = mb + r + hi * 8;
      float vm = validf[c * NQ + qq];
      attF[((size_t)c * NQ + qq) * EMBD + hd * HD + j] = (acc[r] + bv[j]) * vm;
    }
  }
}

// ---------------- K6: camera reduce + count divide ----------------
__global__ void k_reduce(const float* __restrict__ attF,
                         const float* __restrict__ countq,
                         float* __restrict__ slots) {
  int i = blockIdx.x * blockDim.x + threadIdx.x;
  if (i >= NQ * EMBD) return;
  int q = i / EMBD;
  float s = 0.f;
#pragma unroll
  for (int c = 0; c < NCAMS; ++c) s += attF[(size_t)c * NQ * EMBD + i];
  slots[i] = s / countq[q];
}

// ---------------- K7: output GEMM [1600,256]x[256,256] + bias + residual ---
__global__ void k_out(const float* __restrict__ slots,
                      const float* __restrict__ WoutT,
                      const float* __restrict__ bout,
                      const float* __restrict__ query, float* __restrict__ out) {
  int mb = blockIdx.x * 16;
  int nb = blockIdx.y * 16;
  int lane = threadIdx.x;
  int m = lane & 15, hi = lane >> 4, ko = hi * 2, n = lane & 15;
  const float* Arow = &slots[(size_t)(mb + m) * EMBD + ko];
  const float* Brow = &WoutT[(size_t)(nb + n) * EMBD + ko];
  v8f acc = {0.f, 0.f, 0.f, 0.f, 0.f, 0.f, 0.f, 0.f};
  for (int s = 0; s < 64; ++s)
    acc = wmma4(ld2(&Arow[s * 4]), ld2(&Brow[s * 4]), acc);
  int j = nb + n;
#pragma unroll
  for (int r = 0; r < 8; ++r) {
    int q = mb + r + hi * 8;
    out[(size_t)q * EMBD + j] = acc[r] + bout[j] + query[(size_t)q * EMBD + j];
  }
}

extern "C" void kernel_launch(void* const* d_in, const int* in_sizes, int n_in,
                              void* d_out, int out_size, void* d_ws,
                              size_t ws_size, hipStream_t stream) {
  const float* query  = (const float*)d_in[0];
  const float* key    = (const float*)d_in[1];
  const float* value  = (const float*)d_in[2];
  const float* refcam = (const float*)d_in[5];
  const int*   bev    = (const int*)d_in[6];
  const float* Wq   = (const float*)d_in[7];
  const float* bq   = (const float*)d_in[8];
  const float* Wk   = (const float*)d_in[9];
  const float* bk   = (const float*)d_in[10];
  const float* Wv   = (const float*)d_in[11];
  const float* bv   = (const float*)d_in[12];
  const float* Woff = (const float*)d_in[13];
  const float* boff = (const float*)d_in[14];
  const float* Wout = (const float*)d_in[15];
  const float* bout = (const float*)d_in[16];
  float* out = (float*)d_out;

  // workspace carve-up (floats); every region starts at an even float offset
  float* base = (float*)d_ws;
  size_t o = 0;
  float* validf = base + o; o += (size_t)NCAMS * NQ;
  float* countq = base + o; o += NQ;
  int*   q0     = (int*)(base + o); o += 8;
  float* qproj  = base + o; o += (size_t)NCAMS * NQ * NHEAD * HD;
  float* offw   = base + o; o += (size_t)NCAMS * NQ * NHEAD * 64;
  float* kproj  = base + o; o += (size_t)NCAMS * NHEAD * KT * HD;
  float* attw   = base + o; o += (size_t)NCAMS * NQ * NHEAD * KT;
  float* vsum   = base + o; o += (size_t)NCAMS * NQ * NHEAD * HD;
  float* attF   = base + o; o += (size_t)NCAMS * NQ * EMBD;
  float* slots  = base + o; o += (size_t)NQ * EMBD;
  float* WqT    = base + o; o += (size_t)HD * HD;
  float* WkT    = base + o; o += (size_t)HD * HD;
  float* WvT    = base + o; o += (size_t)HD * HD;
  float* WoffT  = base + o; o += (size_t)HD * 64;
  float* WoutT  = base + o; o += (size_t)EMBD * EMBD;

  // weight transposes (tiny)
  k_transpose<<<(HD * HD + 255) / 256, 256, 0, stream>>>(Wq, WqT, HD, HD);
  k_transpose<<<(HD * HD + 255) / 256, 256, 0, stream>>>(Wk, WkT, HD, HD);
  k_transpose<<<(HD * HD + 255) / 256, 256, 0, stream>>>(Wv, WvT, HD, HD);
  k_transpose<<<(HD * 64 + 255) / 256, 256, 0, stream>>>(Woff, WoffT, HD, 64);
  k_transpose<<<(EMBD * EMBD + 255) / 256, 256, 0, stream>>>(Wout, WoutT, EMBD,
                                                             EMBD);

  k_init<<<1, 32, 0, stream>>>(q0);
  k_mask<<<(NQ + 255) / 256, 256, 0, stream>>>(bev, validf, countq, q0);
  k_qoff<<<dim3(NQ / 16, NHEAD, NCAMS), 32, 0, stream>>>(
      query, validf, WqT, bq, WoffT, boff, qproj, offw);
  k_keys<<<dim3(NHEAD, NCAMS), 32, 0, stream>>>(key, refcam, validf, q0, offw,
                                                WkT, bk, kproj);
  k_scores<<<dim3(NQ / 16, NHEAD, NCAMS), 32, 0, stream>>>(qproj, kproj, attw);
  k_vsamp<<<dim3(NQ, NCAMS), 256, 0, stream>>>(value, refcam, validf, offw,
                                               attw, vsum);
  k_vproj<<<dim3(NQ / 16, NHEAD, NCAMS), 32, 0, stream>>>(vsum, WvT, bv, validf,
                                                          attF);
  k_reduce<<<(NQ * EMBD + 255) / 256, 256, 0, stream>>>(attF, countq, slots);
  k_out<<<dim3(NQ / 16, EMBD / 16), 32, 0, stream>>>(slots, WoutT, bout, query,
                                                     out);
}